// ChamferWithNormalLoss_67740224192624
// MI455X (gfx1250) — compile-verified
//
#include <hip/hip_runtime.h>

typedef __attribute__((ext_vector_type(2))) float v2f;
typedef __attribute__((ext_vector_type(4))) float v4f;
typedef __attribute__((ext_vector_type(8))) float v8f;

#define BATCH 8
#define CHN   6
#define NPTS  4096
#define WNORM 0.001f
#define TC    256              // columns per LDS chunk (16 WMMA tiles)
#define NCHUNK (NPTS / TC)     // 16 chunks

// Store an 8-float operand row permuted as [k0,k1,k4,k5,k2,k3,k6,k7] so each
// lane-half's two float2 WMMA operands are one contiguous 16B load.
__device__ __forceinline__ void store_perm8(float* p, float c0, float c1, float c2,
                                            float c3, float c4, float c5, float c6) {
    p[0] = c0; p[1] = c1; p[2] = c4; p[3] = c5;
    p[4] = c2; p[5] = c3; p[6] = c6; p[7] = 0.0f;
}

// ---------------------------------------------------------------------------
// Prep: build rank-7 (padded to 8) GEMM operands for both chamfer passes.
//   Pass1 (min over j for each a-row i):
//     R1[i] = [-2*ap, -w*bn, 1]   C1[j] = [bp, an, bb_j + w]   r1[i] = aa_i
//   Pass2 (min over i for each b-row j):
//     R2[j] = [-2*bp, -w*an, 1]   C2[i] = [ap, bn, aa_i + w]   r2[j] = bb_j
// Input layout is (B, C, N): channel-major over N.
// ---------------------------------------------------------------------------
__global__ void chamfer_prep(const float* __restrict__ a, const float* __restrict__ b,
                             float* __restrict__ R1, float* __restrict__ C1,
                             float* __restrict__ R2, float* __restrict__ C2,
                             float* __restrict__ r1, float* __restrict__ r2) {
    int idx = blockIdx.x * blockDim.x + threadIdx.x;
    if (idx >= BATCH * NPTS) return;
    int bi = idx / NPTS;
    int n  = idx % NPTS;
    const float* ain = a + (size_t)bi * CHN * NPTS + n;
    const float* bin = b + (size_t)bi * CHN * NPTS + n;

    float ap[3], an[3], bp[3], bn[3];
#pragma unroll
    for (int c = 0; c < 3; ++c) {
        ap[c] = ain[(size_t)c * NPTS];
        an[c] = ain[(size_t)(c + 3) * NPTS];
        bp[c] = bin[(size_t)c * NPTS];
        bn[c] = bin[(size_t)(c + 3) * NPTS];
    }
    float aa  = ap[0]*ap[0] + ap[1]*ap[1] + ap[2]*ap[2];
    float bbv = bp[0]*bp[0] + bp[1]*bp[1] + bp[2]*bp[2];

    store_perm8(R1 + (size_t)idx * 8, -2.0f*ap[0], -2.0f*ap[1], -2.0f*ap[2],
                -WNORM*bn[0], -WNORM*bn[1], -WNORM*bn[2], 1.0f);
    store_perm8(C1 + (size_t)idx * 8, bp[0], bp[1], bp[2],
                an[0], an[1], an[2], bbv + WNORM);
    store_perm8(R2 + (size_t)idx * 8, -2.0f*bp[0], -2.0f*bp[1], -2.0f*bp[2],
                -WNORM*an[0], -WNORM*an[1], -WNORM*an[2], 1.0f);
    store_perm8(C2 + (size_t)idx * 8, ap[0], ap[1], ap[2],
                bn[0], bn[1], bn[2], aa + WNORM);

    r1[idx] = aa;
    r2[idx] = bbv;
}

__global__ void chamfer_zero(float* __restrict__ out) {
    if (threadIdx.x == 0 && blockIdx.x == 0) out[0] = 0.0f;
}

// Issue a 32B column-row copy to LDS via two async b128s (IOFFSET is applied
// to both the global and LDS sides per the ISA async pseudocode).
__device__ __forceinline__ void async_copy_row32(unsigned lds_off, const float* gp) {
    asm volatile(
        "global_load_async_to_lds_b128 %0, %1, off\n\t"
        "global_load_async_to_lds_b128 %0, %1, off offset:16"
        :: "v"(lds_off), "v"(gp) : "memory");
}

__device__ __forceinline__ void wait_async0() {
    asm volatile("s_wait_asynccnt 0" ::: "memory");
}

// ---------------------------------------------------------------------------
// Main: 8 waves/block, 16 rows per wave (128 rows/block). Column matrix is
// streamed through a double-buffered 8KB LDS chunk (256 cols) via async-to-LDS.
// Per 16x16 tile: 1x ds_load_b128 + 2x v_wmma_f32_16x16x4_f32 (K=8) +
// 8x v_min_f32 into a persistent min accumulator in WMMA C/D layout.
// ---------------------------------------------------------------------------
__global__ __launch_bounds__(256) void chamfer_wmma(
    const float* __restrict__ R1, const float* __restrict__ C1, const float* __restrict__ r1,
    const float* __restrict__ R2, const float* __restrict__ C2, const float* __restrict__ r2,
    float* __restrict__ out) {

    __shared__ float smem[2 * TC * 8];   // 16 KB double buffer

    const float* R  = blockIdx.z ? R2 : R1;
    const float* Cm = blockIdx.z ? C2 : C1;
    const float* rb = blockIdx.z ? r2 : r1;
    const int batch = blockIdx.y;
    R  += (size_t)batch * NPTS * 8;
    Cm += (size_t)batch * NPTS * 8;
    rb += (size_t)batch * NPTS;

    const int tid  = threadIdx.x;
    const int wave = tid >> 5;
    const int lane = tid & 31;
    const int half = lane >> 4;       // which K-pair this lane carries
    const int l    = lane & 15;
    const int m0   = blockIdx.x * 128 + wave * 16;

    // A operands for this wave's 16 rows: one b128 (permuted row) gives
    // a0 = K{2h,2h+1}, a1 = K{4+2h,4+2h+1}.
    const v4f av = *(const v4f*)(R + (size_t)(m0 + l) * 8 + half * 4);
    const v2f a0 = {av[0], av[1]};
    const v2f a1 = {av[2], av[3]};

    // Each thread owns one column-row (32B) of every chunk.
    const unsigned lds_slot0 = (unsigned)(uintptr_t)(&smem[0]) + (unsigned)tid * 32u;
    const unsigned lds_bufstride = TC * 8 * sizeof(float);
    const float* gcol = Cm + (size_t)tid * 8;

    v8f minacc;
#pragma unroll
    for (int k = 0; k < 8; ++k) minacc[k] = 3.0e38f;

    // Prologue: stage chunk 0.
    async_copy_row32(lds_slot0, gcol);
    wait_async0();
    __syncthreads();

    for (int ch = 0; ch < NCHUNK; ++ch) {
        // Kick off the next chunk into the other buffer while computing.
        if (ch + 1 < NCHUNK) {
            async_copy_row32(lds_slot0 + ((ch + 1) & 1) * lds_bufstride,
                             gcol + (size_t)(ch + 1) * TC * 8);
        }

        const float* buf = smem + (ch & 1) * (TC * 8);
#pragma unroll 4
        for (int t = 0; t < TC / 16; ++t) {
            const v4f bv = *(const v4f*)(buf + (size_t)(t * 16 + l) * 8 + half * 4);
            const v2f b0 = {bv[0], bv[1]};
            const v2f b1 = {bv[2], bv[3]};

            v8f acc = {};
            acc = __builtin_amdgcn_wmma_f32_16x16x4_f32(
                false, a0, false, b0, (short)0, acc, false, false);
            acc = __builtin_amdgcn_wmma_f32_16x16x4_f32(
                false, a1, false, b1, (short)0, acc, false, false);

#pragma unroll
            for (int k = 0; k < 8; ++k) minacc[k] = fminf(minacc[k], acc[k]);
        }

        wait_async0();      // my async writes for chunk ch+1 are done
        __syncthreads();    // everyone done reading chunk ch, chunk ch+1 visible
    }

    // Cross-lane min across the 16 lanes of each half (a tile row lives across
    // lanes 0-15 for M=vgpr, lanes 16-31 for M=vgpr+8), then add row bias.
    float sum = 0.0f;
#pragma unroll
    for (int k = 0; k < 8; ++k) {
        float x = minacc[k];
        x = fminf(x, __shfl_xor(x, 8, 32));
        x = fminf(x, __shfl_xor(x, 4, 32));
        x = fminf(x, __shfl_xor(x, 2, 32));
        x = fminf(x, __shfl_xor(x, 1, 32));
        const int row = m0 + half * 8 + k;
        sum += x + rb[row];
    }
    if (l == 0) atomicAdd(out, sum * (1.0f / (float)BATCH));
}

extern "C" void kernel_launch(void* const* d_in, const int* in_sizes, int n_in,
                              void* d_out, int out_size, void* d_ws, size_t ws_size,
                              hipStream_t stream) {
    const float* a = (const float*)d_in[0];
    const float* b = (const float*)d_in[1];
    float* out = (float*)d_out;
    float* ws  = (float*)d_ws;

    const size_t BN  = (size_t)BATCH * NPTS;
    const size_t BN8 = BN * 8;
    float* R1 = ws;
    float* C1 = ws + BN8;
    float* R2 = ws + 2 * BN8;
    float* C2 = ws + 3 * BN8;
    float* r1 = ws + 4 * BN8;
    float* r2 = ws + 4 * BN8 + BN;

    chamfer_zero<<<1, 64, 0, stream>>>(out);
    chamfer_prep<<<(int)((BN + 255) / 256), 256, 0, stream>>>(a, b, R1, C1, R2, C2, r1, r2);

    dim3 grid(NPTS / 128, BATCH, 2);   // row-blocks x batches x {pass1, pass2}
    chamfer_wmma<<<grid, 256, 0, stream>>>(R1, C1, r1, R2, C2, r2, out);
}